// TemporalAttentionAdj_71055938945735
// MI455X (gfx1250) — compile-verified
//
#include <hip/hip_runtime.h>

typedef float v2f __attribute__((ext_vector_type(2)));
typedef float v8f __attribute__((ext_vector_type(8)));

constexpr int Bc = 16, Nc = 128, Tc = 512, Cc = 64, Hc = 64;

// ---------------------------------------------------------------------------
// Kernel 1: x_mean = mean over N.  x:(B,N,T,C) -> xm:(B,T,C). float4 streaming.
// ---------------------------------------------------------------------------
__global__ void mean_kernel(const float* __restrict__ x, float* __restrict__ xm) {
    int g = blockIdx.x * blockDim.x + threadIdx.x;      // one float4 of (T,C) plane
    const int planeGroups = Tc * Cc / 4;                // 8192 per batch
    int b = g / planeGroups;
    int r = g - b * planeGroups;
    const float* xb = x + (size_t)b * Nc * Tc * Cc + (size_t)r * 4;
    float4 acc = make_float4(0.f, 0.f, 0.f, 0.f);
#pragma unroll 8
    for (int n = 0; n < Nc; ++n) {
        float4 v = *reinterpret_cast<const float4*>(xb + (size_t)n * Tc * Cc);
        acc.x += v.x; acc.y += v.y; acc.z += v.z; acc.w += v.w;
    }
    const float inv = 1.0f / (float)Nc;
    float4 o = make_float4(acc.x * inv, acc.y * inv, acc.z * inv, acc.w * inv);
    *reinterpret_cast<float4*>(xm + (size_t)g * 4) = o;
}

// ---------------------------------------------------------------------------
// Kernel 2: per (b,t) row: proj + bias, LayerNorm over H, L2 normalize.
// 64 threads per block (thread h), LDS broadcast reductions.
// ---------------------------------------------------------------------------
__global__ void qk_kernel(const float* __restrict__ xm,
                          const float* __restrict__ Wq, const float* __restrict__ bq,
                          const float* __restrict__ Wk, const float* __restrict__ bk,
                          float* __restrict__ Q, float* __restrict__ K) {
    __shared__ float sx[Cc];
    __shared__ float sq[Hc];
    __shared__ float sk[Hc];
    const int row = blockIdx.x;          // b*T + t
    const int h = threadIdx.x;           // 0..63
    sx[h] = xm[(size_t)row * Cc + h];
    __syncthreads();

    float aq = bq[h];
    float ak = bk[h];
    const float* wqr = Wq + (size_t)h * Cc;
    const float* wkr = Wk + (size_t)h * Cc;
#pragma unroll
    for (int c = 0; c < Cc; ++c) {
        float xv = sx[c];
        aq = fmaf(xv, wqr[c], aq);
        ak = fmaf(xv, wkr[c], ak);
    }
    sq[h] = aq; sk[h] = ak;
    __syncthreads();

    float mq = 0.f, mk = 0.f;
    for (int i = 0; i < Hc; ++i) { mq += sq[i]; mk += sk[i]; }
    mq *= (1.0f / Hc); mk *= (1.0f / Hc);
    float vq = 0.f, vk = 0.f;
    for (int i = 0; i < Hc; ++i) {
        float dq = sq[i] - mq; vq += dq * dq;
        float dk = sk[i] - mk; vk += dk * dk;
    }
    vq *= (1.0f / Hc); vk *= (1.0f / Hc);

    float zq = (aq - mq) / sqrtf(vq + 1e-5f);
    float zk = (ak - mk) / sqrtf(vk + 1e-5f);
    __syncthreads();
    sq[h] = zq * zq; sk[h] = zk * zk;
    __syncthreads();
    float nq = 0.f, nk = 0.f;
    for (int i = 0; i < Hc; ++i) { nq += sq[i]; nk += sk[i]; }
    nq = sqrtf(nq); nk = sqrtf(nk);

    Q[(size_t)row * Hc + h] = zq / fmaxf(nq, 1e-12f);
    K[(size_t)row * Hc + h] = zk / fmaxf(nk, 1e-12f);
}

// ---------------------------------------------------------------------------
// Kernel 3: logits = clip(Q @ K^T, -6, 6) via V_WMMA_F32_16X16X4_F32.
// One wave per 16x16 output tile; K-dim = 64 -> 16 WMMA steps of k=4.
// A frag: lane m=lane&15 is row M; VGPR0/1 hold K = 2*(lane>>4) + {0,1}.
// B frag mirrors with N=lane (B[k][n] = K[s0+n][k]).
// D: VGPR v -> row t0 + v + 8*(lane>>4), col s0 + (lane&15).
// ---------------------------------------------------------------------------
__global__ void logits_kernel(const float* __restrict__ Q, const float* __restrict__ Kx,
                              float* __restrict__ logits) {
    const int wave = blockIdx.x * (blockDim.x >> 5) + (threadIdx.x >> 5);
    const int lane = threadIdx.x & 31;
    const int b = wave >> 10;                 // / (32*32)
    const int rem = wave & 1023;
    const int t0 = (rem >> 5) << 4;
    const int s0 = (rem & 31) << 4;

    const int m = lane & 15;
    const int koff = (lane >> 4) << 1;
    const float* qrow = Q  + ((size_t)b * Tc + t0 + m) * Hc;
    const float* krow = Kx + ((size_t)b * Tc + s0 + m) * Hc;

    v8f c = {0.f, 0.f, 0.f, 0.f, 0.f, 0.f, 0.f, 0.f};
#pragma unroll
    for (int kk = 0; kk < 16; ++kk) {
        const int k = kk * 4 + koff;
        v2f a  = *reinterpret_cast<const v2f*>(qrow + k);
        v2f bb = *reinterpret_cast<const v2f*>(krow + k);
        c = __builtin_amdgcn_wmma_f32_16x16x4_f32(false, a, false, bb,
                                                  (short)0, c, false, false);
    }

    const int rbase = t0 + ((lane >> 4) << 3);
    const int col = s0 + m;
    float* out = logits + ((size_t)b * Tc + rbase) * Tc + col;
#pragma unroll
    for (int v = 0; v < 8; ++v) {
        float val = c[v];
        val = fminf(fmaxf(val, -6.0f), 6.0f);
        out[(size_t)v * Tc] = val;
    }
}

// ---------------------------------------------------------------------------
// Kernel 4: banded softmax + floor/ceiling + double renormalize -> A.
// One block per (b,t) row. Off-band entries collapse to a single fill value
// (exp(-1e9) underflows to 0 -> floor path). Thread 0 computes the <=2*band
// band values + fill, then all 128 threads stream the 512-wide row.
// ---------------------------------------------------------------------------
__global__ void adj_kernel(const float* __restrict__ logits,
                           const int* __restrict__ band_ptr,
                           float* __restrict__ A) {
    __shared__ float s_band[17];   // offset d+8, supports band <= 8
    __shared__ float s_fill;
    __shared__ int   s_bandw;
    const int r = blockIdx.x;            // b*T + t
    const int t = r & (Tc - 1);
    const float* lrow = logits + (size_t)r * Tc;

    if (threadIdx.x == 0) {
        int band = *band_ptr;
        if (band > 8) band = 8;
        s_bandw = band;
        const int lo = (t - band < 0) ? 0 : t - band;
        const int hi = (t + band > Tc - 1) ? Tc - 1 : t + band;

        float M = -1e30f;
        for (int s = lo; s <= hi; ++s)
            if (s != t) M = fmaxf(M, lrow[s]);

        float p[17];
        float Z = 0.f;
        int nb = 0;
        for (int s = lo; s <= hi; ++s) {
            if (s == t) continue;
            float e = expf(lrow[s] - M);
            p[s - t + 8] = e;
            Z += e;
            ++nb;
        }

        const float f0 = 0.05f / (float)Tc;                  // ROW_FLOOR / T
        float u = fminf(0.95f * 1e-12f + f0, 0.35f);          // off-band value
        float S1 = (float)(Tc - nb) * u;
        for (int s = lo; s <= hi; ++s) {
            if (s == t) continue;
            float a0 = fmaxf(p[s - t + 8] / Z, 1e-12f);
            float v  = fminf(0.95f * a0 + f0, 0.35f);
            p[s - t + 8] = v;
            S1 += v;
        }
        const float S1c = fmaxf(S1, 1e-12f);
        const float fill1 = u / S1c;
        float S2 = (float)(Tc - nb) * fill1;
        for (int s = lo; s <= hi; ++s) {
            if (s == t) continue;
            float a1 = p[s - t + 8] / S1c;
            p[s - t + 8] = a1;
            S2 += a1;
        }
        const float S2c = fmaxf(S2, 1e-12f);
        s_fill = fill1 / S2c;
        for (int s = lo; s <= hi; ++s)
            if (s != t) s_band[s - t + 8] = p[s - t + 8] / S2c;
    }
    __syncthreads();

    const int bandw = s_bandw;
    const float fill = s_fill;
    float* arow = A + (size_t)r * Tc;
    for (int col = threadIdx.x; col < Tc; col += blockDim.x) {
        const int d = col - t;
        float val = fill;
        if (d != 0 && d >= -bandw && d <= bandw) val = s_band[d + 8];
        arow[col] = val;
    }
}

// ---------------------------------------------------------------------------
extern "C" void kernel_launch(void* const* d_in, const int* in_sizes, int n_in,
                              void* d_out, int out_size, void* d_ws, size_t ws_size,
                              hipStream_t stream) {
    const float* x    = (const float*)d_in[0];
    const float* Wq   = (const float*)d_in[1];
    const float* bq   = (const float*)d_in[2];
    const float* Wk   = (const float*)d_in[3];
    const float* bk   = (const float*)d_in[4];
    const int*   band = (const int*)d_in[5];

    float* A      = (float*)d_out;                                  // (B,T,T)
    float* logits = (float*)d_out + (size_t)Bc * Tc * Tc;           // (B,T,T)

    float* xm = (float*)d_ws;                                       // B*T*C
    float* Q  = xm + (size_t)Bc * Tc * Cc;                          // B*T*H
    float* K  = Q  + (size_t)Bc * Tc * Hc;                          // B*T*H

    // 1) mean over N (bandwidth-dominant: 268 MB read)
    {
        int groups = Bc * Tc * Cc / 4;          // 131072
        mean_kernel<<<groups / 256, 256, 0, stream>>>(x, xm);
    }
    // 2) Q/K projection + LayerNorm + L2 normalize
    qk_kernel<<<Bc * Tc, Hc, 0, stream>>>(xm, Wq, bq, Wk, bk, Q, K);
    // 3) logits = clip(Q K^T) via f32 WMMA; 16384 tiles, 4 waves/block
    {
        int tiles = Bc * (Tc / 16) * (Tc / 16); // 16384
        logits_kernel<<<tiles / 4, 128, 0, stream>>>(Q, K, logits);
    }
    // 4) banded softmax -> A
    adj_kernel<<<Bc * Tc, 128, 0, stream>>>(logits, band, A);
}